// HybridSelfAttentionBlockWithDisplacement_35948876267798
// MI455X (gfx1250) — compile-verified
//
#include <hip/hip_runtime.h>
#include <math.h>

typedef __attribute__((ext_vector_type(2))) float v2f;
typedef __attribute__((ext_vector_type(8))) float v8f;

#define BB 8
#define HH 8
#define LL 4096
#define KK 16
#define DD 256
#define DH 64
#define PAD 4
#define ROWSTR (DD + PAD)           // 260 floats: 16B-aligned rows, banks spread by 4
#define EPSF 1e-8f
#define LN_EPSF 1e-5f

__device__ __forceinline__ float sigf(float x) { return 1.f / (1.f + expf(-x)); }

// ---------------- kernel 0: init per-batch min/max slots ----------------
__global__ void k_init_minmax(int* mn, int* mx) {
  int i = threadIdx.x;
  if (i < BB) { mn[i] = 0x7f800000; mx[i] = 0; }   // +inf / 0.0f (std >= 0)
}

// ---------------- kernel 1: per-row feature_std -> per-batch min/max ----
__global__ void k_feature_std(const float* __restrict__ lat,
                              int* __restrict__ mn, int* __restrict__ mx) {
  int row  = blockIdx.x * 8 + (threadIdx.x >> 5);   // one wave per row
  int lane = threadIdx.x & 31;
  const float4* p = (const float4*)(lat + (size_t)row * DD) + lane * 2;
  float s = 0.f, ss = 0.f;
#pragma unroll
  for (int j = 0; j < 2; ++j) {
    float4 v = p[j];
    s  += v.x + v.y + v.z + v.w;
    ss += v.x*v.x + v.y*v.y + v.z*v.z + v.w*v.w;
  }
#pragma unroll
  for (int off = 16; off > 0; off >>= 1) {
    s  += __shfl_xor(s,  off, 32);
    ss += __shfl_xor(ss, off, 32);
  }
  if (lane == 0) {
    float var = (ss - s * s / (float)DD) / (float)(DD - 1);
    float sd  = sqrtf(fmaxf(var, 0.f));
    int b = row / LL;
    atomicMin(mn + b, __float_as_int(sd));
    atomicMax(mx + b, __float_as_int(sd));
  }
}

// ---------------- kernel 2: fused main ----------------------------------
__global__ void __launch_bounds__(128)
k_main(const float* __restrict__ lat,  const float* __restrict__ attn,
       const float* __restrict__ nf,   const float* __restrict__ npos,
       const float* __restrict__ cpos, const float* __restrict__ hw,
       const float* __restrict__ rtemp,const float* __restrict__ lba,
       const float* __restrict__ lbr,  const float* __restrict__ istr,
       const float* __restrict__ w1,   const float* __restrict__ b1,
       const float* __restrict__ lng,  const float* __restrict__ lnb,
       const float* __restrict__ w2,   const float* __restrict__ b2,
       const int* __restrict__ mn,     const int* __restrict__ mx,
       float* __restrict__ out) {
  extern __shared__ float smem[];
  const int tid  = threadIdx.x;
  const int wid  = tid >> 5;
  const int lane = tid & 31;
  const int half = lane >> 4;        // 0: cols 0-127 / h-cols j+0..7 ; 1: cols 128-255 / j+8..15
  const int r16  = lane & 15;        // row within 16-row tile
  float* lat_s = smem + wid * 16 * ROWSTR;

  const int row = (blockIdx.x * 4 + wid) * 16 + r16;   // global (b,l) row
  const int b   = row >> 12;                            // / LL
  const int l   = row & (LL - 1);

  // ---- Phase 1: stream latents (stage to LDS) + neighbor features ----
  float dotk[KK], nfss[KK];
#pragma unroll
  for (int k = 0; k < KK; ++k) { dotk[k] = 0.f; nfss[k] = 0.f; }
  float s = 0.f, ss = 0.f;
  const float* latp = lat + (size_t)row * DD + half * 128;
  const float* nfp  = nf  + (size_t)row * KK * DD + half * 128;
  float* dst = lat_s + r16 * ROWSTR + half * 128;
  for (int i = 0; i < 32; ++i) {
    float4 v = ((const float4*)latp)[i];
    ((float4*)dst)[i] = v;
    s  += v.x + v.y + v.z + v.w;
    ss += v.x*v.x + v.y*v.y + v.z*v.z + v.w*v.w;
#pragma unroll
    for (int k = 0; k < KK; ++k) {
      float4 u = ((const float4*)(nfp + (size_t)k * DD))[i];
      dotk[k] += v.x*u.x + v.y*u.y + v.z*u.z + v.w*u.w;
      nfss[k] += u.x*u.x + u.y*u.y + u.z*u.z + u.w*u.w;
    }
  }
  __syncthreads();   // latents tile visible for WMMA B-operand reads

  // combine lane pair halves
  s  += __shfl_xor(s, 16, 32);
  ss += __shfl_xor(ss, 16, 32);
#pragma unroll
  for (int k = 0; k < KK; ++k) {
    dotk[k] += __shfl_xor(dotk[k], 16, 32);
    nfss[k] += __shfl_xor(nfss[k], 16, 32);
  }

  // ---- Row stats ----
  float lat_n = fmaxf(sqrtf(ss), EPSF);
  float var   = (ss - s * s / (float)DD) / (float)(DD - 1);
  float fstd  = sqrtf(fmaxf(var, 0.f));
  float smn = __int_as_float(mn[b]);
  float smx = __int_as_float(mx[b]);
  float complexity = (fstd - smn) / (smx - smn + EPSF);

  float simk[KK]; float simsum = 0.f;
#pragma unroll
  for (int k = 0; k < KK; ++k) {
    simk[k] = dotk[k] / (lat_n * fmaxf(sqrtf(nfss[k]), EPSF));
    simsum += simk[k];
  }
  float uniq = 1.f - simsum * (1.f / (float)KK);

  // ---- head softmax ----
  float hwv[HH]; float hm = -1e30f;
#pragma unroll
  for (int h = 0; h < HH; ++h) { hwv[h] = hw[h]; hm = fmaxf(hm, hwv[h]); }
  float es = 0.f;
#pragma unroll
  for (int h = 0; h < HH; ++h) { hwv[h] = expf(hwv[h] - hm); es += hwv[h]; }
  float inv_es = 1.f / es;

  // ---- weighted attention: each lane does its 8 k's, pair-share the rest ----
  float wa[8];
#pragma unroll
  for (int kk = 0; kk < 8; ++kk) wa[kk] = 0.f;
  const float* ab = attn + (((size_t)b * HH) * LL + l) * (KK + 1) + 1 + half * 8;
#pragma unroll
  for (int h = 0; h < HH; ++h) {
    const float* ah = ab + (size_t)h * LL * (KK + 1);
#pragma unroll
    for (int kk = 0; kk < 8; ++kk) wa[kk] += hwv[h] * ah[kk];
  }
  float wlo[8], whi[8];
#pragma unroll
  for (int kk = 0; kk < 8; ++kk) {
    float mine  = wa[kk] * inv_es;
    float other = __shfl_xor(wa[kk], 16, 32) * inv_es;
    wlo[kk] = half ? other : mine;
    whi[kk] = half ? mine  : other;
  }

  // ---- geometry: attraction & repulsion (normalizations folded) ----
  float cx = cpos[(size_t)row * 2 + 0];
  float cy = cpos[(size_t)row * 2 + 1];
  float temp = fabsf(rtemp[0]) + EPSF;
  float wsum = 0.f, wcx = 0.f, wcy = 0.f;
  float rsum = 0.f, rpx = 0.f, rpy = 0.f;
#pragma unroll
  for (int k = 0; k < KK; ++k) {
    float wk = (k < 8) ? wlo[k] : whi[k - 8];
    float nx = npos[((size_t)row * KK + k) * 2 + 0];
    float ny = npos[((size_t)row * KK + k) * 2 + 1];
    float dx = nx - cx, dy = ny - cy;
    float dist = sqrtf(dx * dx + dy * dy);
    wsum += wk; wcx += wk * nx; wcy += wk * ny;
    float rs   = expf(simk[k] / temp) / (dist + 0.1f);
    float invd = 1.f / (dist + EPSF);
    rsum += rs; rpx -= rs * dx * invd; rpy -= rs * dy * invd;
  }
  float ax = wcx / (wsum + EPSF) - cx;
  float ay = wcy / (wsum + EPSF) - cy;
  float repx = rpx / (rsum + EPSF);
  float repy = rpy / (rsum + EPSF);

  float importance = 0.5f * complexity + 0.5f * uniq;
  float exsc = 1.f - sigf(istr[0]) * importance;
  float base_a = expf(lba[0]);
  float base_r = expf(lbr[0]);

  // ---- MLP via WMMA f32 16x16x4: D[n][row] = sum_c w1[c][n] * lat[row][c] ----
  v8f acc[4];
#pragma unroll
  for (int t = 0; t < 4; ++t) acc[t] = (v8f)0.f;
  const int k0 = half * 2;
  const float* lrow = lat_s + r16 * ROWSTR;
  for (int c = 0; c < DD; c += 4) {
    v2f bv = *(const v2f*)(lrow + c + k0);           // B: latents, N=row, K=c+k0..c+k0+1
#pragma unroll
    for (int t = 0; t < 4; ++t) {
      int n = t * 16 + r16;                          // A: w1 tile, M=output channel n
      v2f av;
      av.x = w1[(size_t)(c + k0)     * DH + n];
      av.y = w1[(size_t)(c + k0 + 1) * DH + n];
      acc[t] = __builtin_amdgcn_wmma_f32_16x16x4_f32(
          false, av, false, bv, (short)0, acc[t], false, false);
    }
  }

  // lane holds h[row=r16][n = t*16 + half*8 + j]  -> pair covers all 64 cols
  float hs = 0.f, hss = 0.f;
#pragma unroll
  for (int t = 0; t < 4; ++t) {
#pragma unroll
    for (int j = 0; j < 8; ++j) {
      int n = t * 16 + half * 8 + j;
      float x = acc[t][j] + b1[n];
      float g = 0.5f * x * (1.f + erff(x * 0.70710678f));   // exact gelu
      acc[t][j] = g;
      hs += g; hss += g * g;
    }
  }
  hs  += __shfl_xor(hs, 16, 32);
  hss += __shfl_xor(hss, 16, 32);
  float mu   = hs * (1.f / (float)DH);
  float varh = hss * (1.f / (float)DH) - mu * mu;
  float rstd = rsqrtf(varh + LN_EPSF);

  float m0 = 0.f, m1 = 0.f, m2 = 0.f;
#pragma unroll
  for (int t = 0; t < 4; ++t) {
#pragma unroll
    for (int j = 0; j < 8; ++j) {
      int n = t * 16 + half * 8 + j;
      float hn = (acc[t][j] - mu) * rstd * lng[n] + lnb[n];
      m0 += hn * w2[n * 3 + 0];
      m1 += hn * w2[n * 3 + 1];
      m2 += hn * w2[n * 3 + 2];
    }
  }
  m0 += __shfl_xor(m0, 16, 32);
  m1 += __shfl_xor(m1, 16, 32);
  m2 += __shfl_xor(m2, 16, 32);
  m0 += b2[0]; m1 += b2[1]; m2 += b2[2];

  float w_at = base_a * 2.f * sigf(m0);
  float w_rp = base_r * 2.f * sigf(m1);
  float fsc  = exsc * sigf(m2);
  float tw   = w_at + w_rp + EPSF;
  float cox  = (w_at * ax + w_rp * repx) / tw * fsc;
  float coy  = (w_at * ay + w_rp * repy) / tw * fsc;
  float dpx  = tanhf(cox * 0.2f) * 5.f;
  float dpy  = tanhf(coy * 0.2f) * 5.f;

  out[(size_t)row * 2 + half] = half ? dpy : dpx;   // lane writes its component
}

extern "C" void kernel_launch(void* const* d_in, const int* in_sizes, int n_in,
                              void* d_out, int out_size, void* d_ws, size_t ws_size,
                              hipStream_t stream) {
  const float* lat   = (const float*)d_in[0];
  const float* attn  = (const float*)d_in[1];
  const float* nf    = (const float*)d_in[2];
  const float* npos  = (const float*)d_in[3];
  const float* cpos  = (const float*)d_in[4];
  const float* hw    = (const float*)d_in[5];
  const float* rtemp = (const float*)d_in[6];
  const float* lba   = (const float*)d_in[7];
  const float* lbr   = (const float*)d_in[8];
  const float* istr  = (const float*)d_in[9];
  const float* w1    = (const float*)d_in[10];
  const float* b1    = (const float*)d_in[11];
  const float* lng   = (const float*)d_in[12];
  const float* lnb   = (const float*)d_in[13];
  const float* w2    = (const float*)d_in[14];
  const float* b2    = (const float*)d_in[15];
  float* out = (float*)d_out;
  int* mn = (int*)d_ws;
  int* mx = mn + BB;

  k_init_minmax<<<1, 64, 0, stream>>>(mn, mx);
  k_feature_std<<<(BB * LL) / 8, 256, 0, stream>>>(lat, mn, mx);
  size_t shbytes = 4 * 16 * ROWSTR * sizeof(float);
  k_main<<<(BB * LL) / 64, 128, shbytes, stream>>>(
      lat, attn, nf, npos, cpos, hw, rtemp, lba, lbr, istr,
      w1, b1, lng, lnb, w2, b2, mn, mx, out);
}